// TSREC_51041391345941
// MI455X (gfx1250) — compile-verified
//
#include <hip/hip_runtime.h>
#include <hip/hip_bf16.h>

#define N_USERS 50000
#define N_ITEMS 40000
#define N_NODES (N_USERS + N_ITEMS)
#define DIM     128
#define LAYERS  3
#define NNZ_GC  2000000
#define NNZ_IT  800000
#define BATCH   512
#define SLEN    50

typedef __attribute__((ext_vector_type(2))) float v2f;
typedef __attribute__((ext_vector_type(8))) float v8f;

__device__ __forceinline__ void atomAddF(float* p, float v) {
  unsafeAtomicAdd(p, v);   // lowers to global_atomic_add_f32 on gfx1250
}

// ---------------------------------------------------------------------------
// Sparse scatter: out[row] += val * x[col]   (COO segment-sum, atomic scatter)
// One wave32 per nonzero; each lane owns a float4 slice of the 128-dim row.
// Accumulator (<=46MB) is L2-resident on MI455X (192MB L2), so atomics are cheap.
// ---------------------------------------------------------------------------
__global__ void spmm_scatter(const int* __restrict__ rows, const int* __restrict__ cols,
                             const float* __restrict__ vals, const float* __restrict__ x,
                             float* __restrict__ out, int nnz) {
  long long gid = (long long)blockIdx.x * blockDim.x + threadIdx.x;
  int e    = (int)(gid >> 5);
  int lane = (int)(gid & 31);
  if (e >= nnz) return;
  int   r = rows[e];
  int   c = cols[e];
  float v = vals[e];
  const float4 xv = *reinterpret_cast<const float4*>(x + (size_t)c * DIM + lane * 4);
  float* o = out + (size_t)r * DIM + lane * 4;
  atomAddF(o + 0, v * xv.x);
  atomAddF(o + 1, v * xv.y);
  atomAddF(o + 2, v * xv.z);
  atomAddF(o + 3, v * xv.w);
}

// ---------------------------------------------------------------------------
// Elementwise helpers
// ---------------------------------------------------------------------------
__global__ void vec_add(float* __restrict__ acc, const float* __restrict__ v, size_t n) {
  size_t i = (size_t)blockIdx.x * blockDim.x + threadIdx.x;
  if (i < n) acc[i] += v[i];
}

__global__ void vec_scale(float* __restrict__ p, float s, size_t n) {
  size_t i = (size_t)blockIdx.x * blockDim.x + threadIdx.x;
  if (i < n) p[i] *= s;
}

// ---------------------------------------------------------------------------
// acc[row] += x[row] / max(||x[row]||, 1e-12)   — one wave32 per 128-dim row
// ---------------------------------------------------------------------------
__global__ void l2norm_acc(const float* __restrict__ x, float* __restrict__ acc, int rows) {
  long long gid = (long long)blockIdx.x * blockDim.x + threadIdx.x;
  int row  = (int)(gid >> 5);
  int lane = (int)(gid & 31);
  if (row >= rows) return;
  const float4 v = *reinterpret_cast<const float4*>(x + (size_t)row * DIM + lane * 4);
  float ss = v.x * v.x + v.y * v.y + v.z * v.z + v.w * v.w;
  for (int m = 16; m > 0; m >>= 1) ss += __shfl_xor(ss, m, 32);   // wave32 reduction
  float inv = 1.0f / fmaxf(sqrtf(ss), 1e-12f);
  float* a = acc + (size_t)row * DIM + lane * 4;
  a[0] += v.x * inv;
  a[1] += v.y * inv;
  a[2] += v.z * inv;
  a[3] += v.w * inv;
}

// ---------------------------------------------------------------------------
// Session mean pooling: S[b] = (sum_t emb_pad[session_item[b][t]]) / len[b]
// emb_pad row 0 == zeros, rows 1..N_ITEMS == item_out.
// ---------------------------------------------------------------------------
__global__ void sess_mean(const int* __restrict__ sitems, const float* __restrict__ slen,
                          const float* __restrict__ item_out, float* __restrict__ S) {
  int b = blockIdx.x;
  int d = threadIdx.x;                          // 128 threads = 128 dims
  float acc = 0.f;
  for (int t = 0; t < SLEN; ++t) {
    int id = sitems[b * SLEN + t];
    if (t + 1 < SLEN) {                         // prefetch next gathered row
      int idn = sitems[b * SLEN + t + 1];
      if (idn > 0) __builtin_prefetch(item_out + (size_t)(idn - 1) * DIM + d, 0, 0);
    }
    if (id > 0) acc += item_out[(size_t)(id - 1) * DIM + d];
  }
  S[(size_t)b * DIM + d] = acc / slen[b];
}

// ---------------------------------------------------------------------------
// f32 WMMA GEMM:  C[M,N] = A[M,K] @ op(B)     (op = B or B^T, row-major all)
// V_WMMA_F32_16X16X4_F32, exact f32 semantics (matches the f32 reference).
//
// TRANSB is a template parameter so the K-loop carries no runtime branch.
// Each wave computes a 32x16 C tile: two M-accumulators share one B fragment
// per K-step (halves B traffic; interleaved accumulator chains hide the
// WMMA->WMMA RAW latency per ISA 7.12.1). Block = 128 thr = 4 waves = 32x64.
// Requires M%32==0, N%64==0, K%4==0 (true at every call site: M in
// {40000,512}, N in {128,512}, K in {128,512}).
//
// Fragment layouts per CDNA5 ISA 7.12.2:
//   A 16x4 : lane<16 -> row=lane,  VGPR{0,1}=K{0,1};  lane>=16 -> K{2,3}
//   B 4x16 : lane<16 -> col=lane,  VGPR{0,1}=K{0,1};  lane>=16 -> K{2,3}
//   C 16x16: VGPR r  : lane<16 -> (M=r, N=lane); lane>=16 -> (M=r+8, N=lane-16)
// ---------------------------------------------------------------------------
template <int TRANSB>
__global__ __launch_bounds__(128)
void wmma_gemm_f32(const float* __restrict__ A, const float* __restrict__ B,
                   float* __restrict__ C, int M, int N, int K,
                   int lda, int ldb, int ldc) {
  const int lane  = threadIdx.x & 31;
  const int wid   = threadIdx.x >> 5;
  const int tn    = blockIdx.x * 64 + wid * 16;
  const int tm    = blockIdx.y * 32;
  if (tm >= M || tn >= N) return;               // wave-uniform: EXEC stays all-ones
  const int idx15 = lane & 15;                  // A-row / B-col owned by this lane
  const int khalf = lane >> 4;                  // 0 -> K{0,1}, 1 -> K{2,3}

  v8f c0 = {};
  v8f c1 = {};
  const float* arow0 = A + (size_t)(tm + idx15) * lda;
  const float* arow1 = arow0 + (size_t)16 * lda;
  const float* bbase = TRANSB ? (B + (size_t)(tn + idx15) * ldb)   // B[n][k]
                              : (B + tn + idx15);                  // B[k][n]

#pragma unroll 4
  for (int k = 0; k < K; k += 4) {
    const int k0 = k + 2 * khalf;
    v2f a0, a1, b;
    a0.x = arow0[k0];
    a0.y = arow0[k0 + 1];
    a1.x = arow1[k0];
    a1.y = arow1[k0 + 1];
    if (TRANSB) {
      b.x = bbase[k0];
      b.y = bbase[k0 + 1];
    } else {
      b.x = bbase[(size_t)k0 * ldb];
      b.y = bbase[(size_t)(k0 + 1) * ldb];
    }
    c0 = __builtin_amdgcn_wmma_f32_16x16x4_f32(
        false, a0, false, b, (short)0, c0, false, false);
    c1 = __builtin_amdgcn_wmma_f32_16x16x4_f32(
        false, a1, false, b, (short)0, c1, false, false);
  }
#pragma unroll
  for (int r = 0; r < 8; ++r) {
    const int row = tm + r + 8 * khalf;
    C[(size_t)row * ldc + tn + idx15]        = c0[r];
    C[(size_t)(row + 16) * ldc + tn + idx15] = c1[r];
  }
}

// ---------------------------------------------------------------------------
// Host driver
// ---------------------------------------------------------------------------
extern "C" void kernel_launch(void* const* d_in, const int* in_sizes, int n_in,
                              void* d_out, int out_size, void* d_ws, size_t ws_size,
                              hipStream_t stream) {
  (void)in_sizes; (void)n_in; (void)out_size; (void)ws_size;

  const float* ego       = (const float*)d_in[0];
  const float* gc_vals   = (const float*)d_in[1];
  const float* item_emb  = (const float*)d_in[2];
  const float* item_vals = (const float*)d_in[3];
  const float* w_item    = (const float*)d_in[4];
  const float* w_sess    = (const float*)d_in[5];
  const float* slen      = (const float*)d_in[6];
  const float* D_mat     = (const float*)d_in[7];
  const float* A_mat     = (const float*)d_in[8];
  const int*   gc_idx    = (const int*)d_in[9];
  const int*   item_idx  = (const int*)d_in[10];
  const int*   sitems    = (const int*)d_in[11];

  float* out = (float*)d_out;
  const size_t NN = (size_t)N_NODES * DIM;   // 11,520,000
  const size_t NI = (size_t)N_ITEMS * DIM;   //  5,120,000
  const size_t SN = (size_t)BATCH * DIM;     //     65,536

  // Output layout doubles as accumulators: [chain1 NN][chain2 NN][item NI][sess SN]
  float* acc1 = out;
  float* acc2 = out + NN;
  float* iacc = out + 2 * NN;
  float* sacc = out + 2 * NN + NI;

  // Workspace
  float* V  = (float*)d_ws;   // current node embedding
  float* T  = V + NN;         // spmm target / ping-pong
  float* E  = T + NN;         // current item embedding
  float* ET = E + NI;         // gemm temp
  float* S  = ET + NI;        // current session embedding
  float* ST = S + SN;         // gemm temp
  float* DA = ST + SN;        // 512x512 = D @ A

  auto launch_spmm = [&](const int* idx2, int nnz, const float* vals,
                         const float* x, float* o) {
    long long thr = (long long)nnz * 32;
    int blocks = (int)((thr + 255) / 256);
    spmm_scatter<<<blocks, 256, 0, stream>>>(idx2, idx2 + nnz, vals, x, o, nnz);
  };
  auto launch_gemm = [&](const float* A, const float* B, float* C,
                         int M, int N, int K, int lda, int ldb, int ldc, int tB) {
    dim3 grid(N / 64, M / 32);
    if (tB)
      wmma_gemm_f32<1><<<grid, 128, 0, stream>>>(A, B, C, M, N, K, lda, ldb, ldc);
    else
      wmma_gemm_f32<0><<<grid, 128, 0, stream>>>(A, B, C, M, N, K, lda, ldb, ldc);
  };
  auto launch_add = [&](float* a, const float* v, size_t n) {
    vec_add<<<(int)((n + 255) / 256), 256, 0, stream>>>(a, v, n);
  };
  auto launch_scale = [&](float* p, float s, size_t n) {
    vec_scale<<<(int)((n + 255) / 256), 256, 0, stream>>>(p, s, n);
  };
  auto launch_l2acc = [&](const float* x, float* a, int rows) {
    l2norm_acc<<<(rows + 7) / 8, 256, 0, stream>>>(x, a, rows);  // 8 waves/block
  };

  // ----- Graph-conv chains (two views) -----------------------------------
  for (int view = 0; view < 2; ++view) {
    float* chain_acc = view == 0 ? acc1 : acc2;
    hipMemcpyAsync(V, ego, NN * sizeof(float), hipMemcpyDeviceToDevice, stream);
    hipMemcpyAsync(chain_acc, ego, NN * sizeof(float), hipMemcpyDeviceToDevice, stream);
    float* cur = V;
    float* nxt = T;
    for (int l = 0; l < LAYERS; ++l) {
      hipMemsetAsync(nxt, 0, NN * sizeof(float), stream);
      const float* vals = gc_vals + ((size_t)l * 2 + view) * NNZ_GC;
      launch_spmm(gc_idx, NNZ_GC, vals, cur, nxt);
      launch_add(chain_acc, nxt, NN);
      float* tmp = cur; cur = nxt; nxt = tmp;
    }
  }

  // ----- Item chain:  e = l2norm(spmm(e @ W_l^T)) accumulation -----------
  hipMemcpyAsync(E, item_emb, NI * sizeof(float), hipMemcpyDeviceToDevice, stream);
  hipMemcpyAsync(iacc, item_emb, NI * sizeof(float), hipMemcpyDeviceToDevice, stream);
  for (int l = 0; l < LAYERS; ++l) {
    // ET = E @ w_item[l]^T   (40000x128 @ 128x128^T, f32 WMMA)
    launch_gemm(E, w_item + (size_t)l * DIM * DIM, ET,
                N_ITEMS, DIM, DIM, DIM, DIM, DIM, /*transB=*/1);
    hipMemsetAsync(E, 0, NI * sizeof(float), stream);
    launch_spmm(item_idx, NNZ_IT, item_vals, ET, E);
    launch_l2acc(E, iacc, N_ITEMS);
  }
  launch_scale(iacc, 0.25f, NI);                // item_out finalized (needed below)

  // ----- Session branch ---------------------------------------------------
  sess_mean<<<BATCH, DIM, 0, stream>>>(sitems, slen, iacc, S);
  hipMemcpyAsync(sacc, S, SN * sizeof(float), hipMemcpyDeviceToDevice, stream);
  // DA = D @ A  (512x512x512, f32 WMMA)
  launch_gemm(D_mat, A_mat, DA, BATCH, BATCH, BATCH, BATCH, BATCH, BATCH, 0);
  for (int l = 0; l < LAYERS; ++l) {
    // ST = S @ w_sess[l]^T   (512x128x128)
    launch_gemm(S, w_sess + (size_t)l * DIM * DIM, ST,
                BATCH, DIM, DIM, DIM, DIM, DIM, /*transB=*/1);
    // S = DA @ ST            (512x128x512)
    launch_gemm(DA, ST, S, BATCH, DIM, BATCH, BATCH, DIM, DIM, /*transB=*/0);
    launch_l2acc(S, sacc, BATCH);
  }
  launch_scale(sacc, 0.25f, SN);

  // ----- Finalize chain outputs (acc1/acc2 are contiguous in d_out) ------
  launch_scale(acc1, 0.25f, 2 * NN);
}